// spikeLayer_2576980377763
// MI455X (gfx1250) — compile-verified
//
#include <hip/hip_runtime.h>

// ---------------------------------------------------------------------------
// SLAYER spike layer for gfx1250 (MI455X, wave32):
//   Kernel 1: PSP conv as banded-Toeplitz GEMM using V_WMMA_F32_16X16X4_F32.
//             Branch-free interior path (b64 A-loads, batched WMMA), guarded
//             path only for the final partial column tile.
//   Kernel 2: sequential refractory scan, LDS-tiled for coalescing.
// ---------------------------------------------------------------------------

#define THETA    10.0f
#define T_SAMPLE 300

typedef __attribute__((ext_vector_type(2))) float v2f;
typedef __attribute__((ext_vector_type(8))) float v8f;

#define MAX_BAND_ROWS 128   // >= round16(Ksrm-1 + 16) ; Ksrm=77 -> 96/112 rows
#define MAXT 15             // >= Kref-1 (Kref = 11 -> 10 live entries)

// One 16-deep k-tile, no bounds checks (interior column tiles).
__device__ __forceinline__ v8f ktile_full(const float* __restrict__ arow,
                                          const float* wlds,
                                          int kt, int krel, int row, int kp,
                                          v8f c)
{
    v2f a[4], b[4];
#pragma unroll
    for (int j = 0; j < 4; ++j)               // aligned 8B loads (ka even,
        a[j] = *reinterpret_cast<const v2f*>(arow + kt + 4 * j + kp); // row stride 1200B)
#pragma unroll
    for (int j = 0; j < 4; ++j) {
        const int rb = krel + 4 * j + kp;
        b[j].x = wlds[rb * 16 + row];
        b[j].y = wlds[(rb + 1) * 16 + row];
    }
#pragma unroll
    for (int j = 0; j < 4; ++j)
        c = __builtin_amdgcn_wmma_f32_16x16x4_f32(
                false, a[j], false, b[j], (short)0, c, false, false);
    return c;
}

// One 16-deep k-tile with per-element bounds checks (last column tile only).
__device__ __forceinline__ v8f ktile_guard(const float* __restrict__ arow,
                                           const float* wlds,
                                           int kt, int krel, int row, int kp,
                                           int T, v8f c)
{
    v2f a[4], b[4];
#pragma unroll
    for (int j = 0; j < 4; ++j) {
        const int ka = kt + 4 * j + kp;
        a[j].x = (ka     < T) ? arow[ka]     : 0.0f;
        a[j].y = (ka + 1 < T) ? arow[ka + 1] : 0.0f;
    }
#pragma unroll
    for (int j = 0; j < 4; ++j) {
        const int rb = krel + 4 * j + kp;
        b[j].x = wlds[rb * 16 + row];
        b[j].y = wlds[(rb + 1) * 16 + row];
    }
#pragma unroll
    for (int j = 0; j < 4; ++j)
        c = __builtin_amdgcn_wmma_f32_16x16x4_f32(
                false, a[j], false, b[j], (short)0, c, false, false);
    return c;
}

// ---------------------------------------------------------------------------
// Kernel 1: u[b,t] = sum_k spike[b,k] * f[t-k], 0 <= t-k < Ksrm.
// Block = 256 threads = 8 waves. Wave w computes the 16x16 tile at
// (m0 = 128*blockIdx.x + 16*w, t0 = 16*blockIdx.y). The Toeplitz band
// W[k, t0..t0+15] is staged in LDS once per block and shared by all waves.
// ---------------------------------------------------------------------------
__global__ __launch_bounds__(256) void psp_wmma_kernel(
    const float* __restrict__ S,    // (B, T) spikes
    const float* __restrict__ srm,  // (Ksrm) alpha kernel
    float*       __restrict__ U,    // (B, T) membrane potential out
    int B, int T, int Ksrm)
{
    __shared__ float wlds[MAX_BAND_ROWS * 16];

    const int tid  = threadIdx.x;
    const int lane = tid & 31;
    const int wave = tid >> 5;
    const int t0   = blockIdx.y * 16;

    // k-band for this column tile: k in [t0-(Ksrm-1), t0+15], aligned down to 16.
    int kmin     = t0 - (Ksrm - 1);
    int kt_start = (kmin <= 0) ? 0 : (kmin & ~15);
    int lo       = t0 + 16 - MAX_BAND_ROWS;          // LDS capacity guard
    if (lo > 0 && kt_start < lo) kt_start = (lo + 15) & ~15;
    const int rows = (t0 + 16) - kt_start;           // multiple of 16, <= 128

    // Stage Toeplitz band into LDS: wlds[r*16 + n] = f[(t0+n) - (kt_start+r)]
    for (int i = tid; i < rows * 16; i += 256) {
        int r = i >> 4, n = i & 15;
        int d = (t0 + n) - (kt_start + r);
        wlds[i] = (d >= 0 && d < Ksrm) ? srm[d] : 0.0f;
    }
    __syncthreads();

    const int m0  = blockIdx.x * 128 + wave * 16;
    const int row = lane & 15;            // M for A-frag, N for B-frag
    const int kp  = (lane >> 4) * 2;      // K-pair select (lanes 16-31 -> +2)
    const float* arow = S + (size_t)(m0 + row) * T;

    v8f c = {};                           // 16x16 f32 accumulator tile
    if (t0 + 16 <= T) {
        // Interior tiles: fully in-bounds, branch-free loads + batched WMMA.
        for (int kt = kt_start; kt < t0 + 16; kt += 16) {
            // WGP-scope prefetch of the next k-tile (locality 3 -> near cache).
            __builtin_prefetch(arow + kt + 16 + kp, 0, 3);
            c = ktile_full(arow, wlds, kt, kt - kt_start, row, kp, c);
        }
    } else {
        // Last column tile: per-element guards.
        for (int kt = kt_start; kt < t0 + 16; kt += 16)
            c = ktile_guard(arow, wlds, kt, kt - kt_start, row, kp, T, c);
    }

    // C/D layout: VGPR g, lanes 0-15 -> (M=g, N=lane); lanes 16-31 -> (M=8+g).
    const int col = t0 + row;
    if (col < T) {
        const int mbase = m0 + (lane >> 4) * 8;
#pragma unroll
        for (int g = 0; g < 8; ++g)
            U[(size_t)(mbase + g) * T + col] = c[g];
    }
}

// ---------------------------------------------------------------------------
// Kernel 2: refractory scan. One lane per neuron; pending buffer + ref tail
// in registers (zero-padded to MAXT). 32x32 tiles staged through LDS
// (stride 33 -> bank-conflict free on 64 banks) so global traffic stays
// coalesced. Operates in place on U (reads u, writes spikes).
// ---------------------------------------------------------------------------
__global__ __launch_bounds__(256) void spike_scan_kernel(
    float*       __restrict__ US,    // in: u (B,T) ; out: spikes (B,T)
    const float* __restrict__ refk,  // (Kref) refractory kernel, refk[0]==0
    int B, int T, int Kref)
{
    __shared__ float tile[8][32 * 33];

    const int tid  = threadIdx.x;
    const int lane = tid & 31;
    const int wave = tid >> 5;
    float* lds = tile[wave];
    const int b0 = blockIdx.x * 256 + wave * 32;     // this wave's 32 neurons

    float rt[MAXT];
#pragma unroll
    for (int i = 0; i < MAXT; ++i)
        rt[i] = (i + 1 < Kref) ? refk[i + 1] : 0.0f; // ref_tail, zero padded

    float pend[MAXT];
#pragma unroll
    for (int i = 0; i < MAXT; ++i) pend[i] = 0.0f;

    for (int tc = 0; tc < T; tc += 32) {
        int nT = T - tc; if (nT > 32) nT = 32;

        // Coalesced load of (32 neurons x nT times) into transposed LDS tile.
        if (lane < nT) {
#pragma unroll
            for (int j = 0; j < 32; ++j)
                lds[j * 33 + lane] = US[(size_t)(b0 + j) * T + tc + lane];
        }
        __syncthreads();

        // Sequential scan: lane owns neuron b0+lane, walks its LDS row.
        for (int tt = 0; tt < nT; ++tt) {
            float ueff = lds[lane * 33 + tt] + pend[0];
            float sf   = (ueff >= THETA) ? 1.0f : 0.0f;   // spike, 1/Ts == 1
#pragma unroll
            for (int i = 0; i < MAXT - 1; ++i)
                pend[i] = fmaf(sf, rt[i], pend[i + 1]);   // shift + inject
            pend[MAXT - 1] = sf * rt[MAXT - 1];
            lds[lane * 33 + tt] = sf;
        }
        __syncthreads();

        // Coalesced store of spikes back to global (in place).
        if (lane < nT) {
#pragma unroll
            for (int j = 0; j < 32; ++j)
                US[(size_t)(b0 + j) * T + tc + lane] = lds[j * 33 + lane];
        }
        __syncthreads();
    }
}

// ---------------------------------------------------------------------------
extern "C" void kernel_launch(void* const* d_in, const int* in_sizes, int n_in,
                              void* d_out, int out_size, void* d_ws, size_t ws_size,
                              hipStream_t stream)
{
    const float* spike = (const float*)d_in[0];   // (N,C,H,W,T) f32, flat (B,T)
    const float* srm   = (const float*)d_in[1];   // (Ksrm) f32
    const float* refk  = (const float*)d_in[2];   // (Kref) f32
    float* out = (float*)d_out;                   // (B,T) f32 spikes

    const int T    = T_SAMPLE;
    const int B    = in_sizes[0] / T;             // 65536
    const int Ksrm = in_sizes[1];                 // ~77
    const int Kref = in_sizes[2];                 // ~11

    // Phase 1: PSP conv as WMMA GEMM. u -> d_out (used as intermediate).
    dim3 g1(B / 128, (T + 15) / 16);
    psp_wmma_kernel<<<g1, dim3(256), 0, stream>>>(spike, srm, out, B, T, Ksrm);

    // Phase 2: refractory scan in place on d_out.
    dim3 g2(B / 256);
    spike_scan_kernel<<<g2, dim3(256), 0, stream>>>(out, refk, B, T, Kref);
}